// Edgeconvmodel_85418309582937
// MI455X (gfx1250) — compile-verified
//
#include <hip/hip_runtime.h>
#include <hip/hip_bf16.h>
#include <math.h>

// ---------------------------------------------------------------------------
// EdgeConv (t = T-1 slice only) + LSTM + linear head for gfx1250 (MI455X).
// Matrix work runs on V_WMMA_F32_16X16X4_F32 (exact f32 D = A*B + C).
// All WMMA operands staged through LDS; hot loop is branchless.
// ---------------------------------------------------------------------------

typedef __attribute__((ext_vector_type(2))) float v2f;
typedef __attribute__((ext_vector_type(8))) float v8f;

#define NG    8      // graphs kept (b = 0..7, t = 11)
#define NN    500    // nodes per graph
#define KNN   20
#define TSTEP 11     // t index used by the output
#define TT    12

// ------------------------- feature gather (t = 11) -------------------------
__global__ __launch_bounds__(64)
void gather_feats_kernel(const float* __restrict__ x,
                         const float* __restrict__ lat,
                         const float* __restrict__ lon,
                         float* __restrict__ feats0) {  // [NG][NN][3]
  int t = blockIdx.x * 64 + threadIdx.x;
  if (t >= NG * NN) return;
  int b = t / NN, i = t % NN;
  size_t src = (size_t)b * TT * NN + (size_t)TSTEP * NN + i;
  feats0[(size_t)t * 3 + 0] = x[src];
  feats0[(size_t)t * 3 + 1] = lat[src];
  feats0[(size_t)t * 3 + 2] = lon[src];
}

// ------------------------------ kNN (top-20) -------------------------------
// One thread per (graph, node). d = |xi|^2 + |xj|^2 - 2<xi,xj> (same formula
// as the reference). Ascending-j scan + strict-less insertion reproduces
// top_k's lower-index tie-break.
template <int F>
__global__ __launch_bounds__(64)
void knn_kernel(const float* __restrict__ feats,  // [NG][NN][F]
                int* __restrict__ knn) {          // [NG][NN][KNN]
  int t = blockIdx.x * 64 + threadIdx.x;
  if (t >= NG * NN) return;
  int g = t / NN, i = t % NN;
  const float* fg = feats + (size_t)g * NN * F;

  float xi[F];
  float sqi = 0.f;
#pragma unroll
  for (int d = 0; d < F; ++d) { xi[d] = fg[(size_t)i * F + d]; sqi += xi[d] * xi[d]; }

  float bd[KNN];
  int   bi[KNN];
#pragma unroll
  for (int q = 0; q < KNN; ++q) { bd[q] = 3.402823466e38f; bi[q] = 0x7fffffff; }

  for (int j = 0; j < NN; ++j) {
    const float* xj = fg + (size_t)j * F;
    float dot = 0.f, sqj = 0.f;
#pragma unroll
    for (int d = 0; d < F; ++d) { float v = xj[d]; dot += xi[d] * v; sqj += v * v; }
    float dist = sqi + sqj - 2.f * dot;
    if (dist < bd[KNN - 1]) {
      bd[KNN - 1] = dist; bi[KNN - 1] = j;
#pragma unroll
      for (int q = KNN - 1; q > 0; --q) {
        if (bd[q] < bd[q - 1]) {
          float td = bd[q]; bd[q] = bd[q - 1]; bd[q - 1] = td;
          int   ti = bi[q]; bi[q] = bi[q - 1]; bi[q - 1] = ti;
        }
      }
    }
  }
  int* o = knn + (size_t)t * KNN;
#pragma unroll
  for (int q = 0; q < KNN; ++q) o[q] = bi[q];
}

// ------------------------- EdgeConv MLP via WMMA ---------------------------
// One wave per 16-node tile. Weights / biases zero-padded to WMMA tile sizes
// in LDS; edge-feature tile [xi | xj-xi] staged in LDS (xi half written once,
// neighbor half refreshed per k). For each of the 20 neighbors:
//   GEMM1 (A=edge feat 16x2F, B=W1 2FxH1) -> +b1 -> leaky_relu -> LDS
//   GEMM2 (A=y1 16xH1 from LDS, B=W2 H1xH2) -> +b2 -> running max (C layout)
template <int F, int H1, int H2>
__global__ __launch_bounds__(32)
void edgeconv_kernel(const float* __restrict__ feats,  // [NG][NN][F]
                     const int*   __restrict__ knn,    // [NG][NN][KNN]
                     const float* __restrict__ W1,     // [2F][H1]
                     const float* __restrict__ b1,     // [H1]
                     const float* __restrict__ W2,     // [H1][H2]
                     const float* __restrict__ b2,     // [H2]
                     float* __restrict__ out) {        // [NG][NN][H2]
  constexpr int TwoF = 2 * F;
  constexpr int KC1  = (TwoF + 3) / 4;   // K chunks, GEMM1
  constexpr int K1P  = KC1 * 4;          // padded K
  constexpr int NT1  = (H1 + 15) / 16;   // N tiles, GEMM1
  constexpr int N1P  = NT1 * 16;         // padded N
  constexpr int KC2  = H1 / 4;           // H1 always a multiple of 4
  constexpr int NT2  = (H2 + 15) / 16;   // N tiles, GEMM2
  constexpr int N2P  = NT2 * 16;

  const int g     = blockIdx.y;
  const int node0 = blockIdx.x * 16;
  const int lane  = threadIdx.x;       // 0..31
  const int mrow  = lane & 15;         // A-matrix row (M)
  const int khalf = (lane >> 4) * 2;   // A/B K sub-offset: 0 or 2
  const int ncol  = lane & 15;         // B/C column within tile
  const int crow0 = (lane >> 4) * 8;   // C-matrix row base for this lane half

  __shared__ float w1s[K1P][N1P];
  __shared__ float w2s[H1][N2P];
  __shared__ float b1s[N1P];
  __shared__ float b2s[N2P];
  __shared__ float xis[16][F];
  __shared__ float fes[16][K1P];
  __shared__ float y1s[16][N1P];
  __shared__ int   nbrs[16];

  const float* fg = feats + (size_t)g * NN * F;

  // ---- stage zero-padded weights & biases (once per block) ----
  for (int idx = lane; idx < K1P * N1P; idx += 32) {
    int rr = idx / N1P, cc = idx % N1P;
    w1s[rr][cc] = (rr < TwoF && cc < H1) ? W1[(size_t)rr * H1 + cc] : 0.f;
  }
  for (int idx = lane; idx < H1 * N2P; idx += 32) {
    int rr = idx / N2P, cc = idx % N2P;
    w2s[rr][cc] = (cc < H2) ? W2[(size_t)rr * H2 + cc] : 0.f;
  }
  for (int idx = lane; idx < N1P; idx += 32) b1s[idx] = (idx < H1) ? b1[idx] : 0.f;
  for (int idx = lane; idx < N2P; idx += 32) b2s[idx] = (idx < H2) ? b2[idx] : 0.f;
  for (int idx = lane; idx < 16 * K1P; idx += 32) fes[idx / K1P][idx % K1P] = 0.f;
  __syncthreads();

  // ---- stage xi tile; xi is also the (loop-invariant) first feat half ----
  for (int idx = lane; idx < 16 * F; idx += 32) {
    int m = idx / F, d = idx % F;
    int node = node0 + m; if (node > NN - 1) node = NN - 1;
    float v = fg[(size_t)node * F + d];
    xis[m][d] = v;
    fes[m][d] = v;
  }

  v8f mx[NT2];
#pragma unroll
  for (int nt = 0; nt < NT2; ++nt)
#pragma unroll
    for (int r = 0; r < 8; ++r) mx[nt][r] = -3.402823466e38f;

  for (int k = 0; k < KNN; ++k) {
    if (lane < 16) {
      int node = node0 + lane; if (node > NN - 1) node = NN - 1;
      nbrs[lane] = knn[((size_t)g * NN + node) * KNN + k];
    }
    __syncthreads();
    // refresh the xj - xi half of the edge-feature tile (coalesced)
    for (int idx = lane; idx < 16 * F; idx += 32) {
      int m = idx / F, d = idx % F;
      fes[m][F + d] = fg[(size_t)nbrs[m] * F + d] - xis[m][d];
    }
    __syncthreads();

    // A fragments for GEMM1 (reused across all N tiles)
    v2f afrag[KC1];
#pragma unroll
    for (int kk = 0; kk < KC1; ++kk) {
      int d0 = kk * 4 + khalf;
      afrag[kk].x = fes[mrow][d0];
      afrag[kk].y = fes[mrow][d0 + 1];
    }

    // ---- GEMM1 + bias + leaky_relu -> LDS (fully branchless) ----
#pragma unroll
    for (int nt = 0; nt < NT1; ++nt) {
      const int col = nt * 16 + ncol;
      v8f acc;
#pragma unroll
      for (int r = 0; r < 8; ++r) acc[r] = 0.f;
#pragma unroll
      for (int kk = 0; kk < KC1; ++kk) {
        int d0 = kk * 4 + khalf;
        v2f bf; bf.x = w1s[d0][col]; bf.y = w1s[d0 + 1][col];
        acc = __builtin_amdgcn_wmma_f32_16x16x4_f32(false, afrag[kk],
                                                    false, bf,
                                                    (short)0, acc, false, false);
      }
      float bias = b1s[col];
#pragma unroll
      for (int r = 0; r < 8; ++r) {
        float v = acc[r] + bias;
        v = (v > 0.f) ? v : 0.01f * v;      // leaky_relu, slope 0.01
        y1s[crow0 + r][col] = v;
      }
    }
    __syncthreads();

    // A fragments for GEMM2
    v2f a2[KC2];
#pragma unroll
    for (int kk = 0; kk < KC2; ++kk) {
      int d0 = kk * 4 + khalf;
      a2[kk].x = y1s[mrow][d0];
      a2[kk].y = y1s[mrow][d0 + 1];
    }

    // ---- GEMM2 + bias -> running max (fully branchless) ----
#pragma unroll
    for (int nt = 0; nt < NT2; ++nt) {
      const int col = nt * 16 + ncol;
      v8f acc;
#pragma unroll
      for (int r = 0; r < 8; ++r) acc[r] = 0.f;
#pragma unroll
      for (int kk = 0; kk < KC2; ++kk) {
        int d0 = kk * 4 + khalf;
        v2f bf; bf.x = w2s[d0][col]; bf.y = w2s[d0 + 1][col];
        acc = __builtin_amdgcn_wmma_f32_16x16x4_f32(false, a2[kk],
                                                    false, bf,
                                                    (short)0, acc, false, false);
      }
      float bias = b2s[col];
#pragma unroll
      for (int r = 0; r < 8; ++r) mx[nt][r] = fmaxf(mx[nt][r], acc[r] + bias);
    }
    __syncthreads();
  }

  // store max-pooled features (masks only here)
#pragma unroll
  for (int nt = 0; nt < NT2; ++nt) {
    const int col = nt * 16 + ncol;
#pragma unroll
    for (int r = 0; r < 8; ++r) {
      int node = node0 + crow0 + r;
      if (node < NN && col < H2)
        out[((size_t)g * NN + node) * H2 + col] = mx[nt][r];
    }
  }
}

// ----------------------------- LSTM (serial) -------------------------------
// 4000-step recurrence, x in R^8, h in R^32. One wave32, lane r owns gate
// rows r, 32+r, 64+r, 96+r; h broadcast by __shfl. No LDS, no barriers.
__device__ __forceinline__ float sigm(float v) { return 1.f / (1.f + __expf(-v)); }

__global__ __launch_bounds__(32)
void lstm_kernel(const float* __restrict__ h3,    // [NG*NN][8]
                 const float* __restrict__ Wih,   // [128][8]
                 const float* __restrict__ Whh,   // [128][32]
                 const float* __restrict__ bih,   // [128]
                 const float* __restrict__ bhh,   // [128]
                 float* __restrict__ outh) {      // [NG*NN][32]
  const int r = threadIdx.x;

  float wih[4][8], whh[4][32], bsum[4];
#pragma unroll
  for (int q = 0; q < 4; ++q) {
    int row = q * 32 + r;
#pragma unroll
    for (int f = 0; f < 8; ++f)  wih[q][f] = Wih[(size_t)row * 8 + f];
#pragma unroll
    for (int j = 0; j < 32; ++j) whh[q][j] = Whh[(size_t)row * 32 + j];
    bsum[q] = bih[row] + bhh[row];
  }

  float h = 0.f, c = 0.f;
  for (int n = 0; n < NG * NN; ++n) {
    float xf[8];
#pragma unroll
    for (int f = 0; f < 8; ++f) xf[f] = h3[(size_t)n * 8 + f];

    float acc0 = bsum[0], acc1 = bsum[1], acc2 = bsum[2], acc3 = bsum[3];
#pragma unroll
    for (int f = 0; f < 8; ++f) {
      acc0 += wih[0][f] * xf[f]; acc1 += wih[1][f] * xf[f];
      acc2 += wih[2][f] * xf[f]; acc3 += wih[3][f] * xf[f];
    }
#pragma unroll
    for (int j = 0; j < 32; ++j) {
      float hv = __shfl(h, j, 32);
      acc0 += whh[0][j] * hv; acc1 += whh[1][j] * hv;
      acc2 += whh[2][j] * hv; acc3 += whh[3][j] * hv;
    }
    // split order: i, f, g, o
    float ig = sigm(acc0), fg = sigm(acc1), gg = tanhf(acc2), og = sigm(acc3);
    c = fg * c + ig * gg;
    h = og * tanhf(c);
    outh[(size_t)n * 32 + r] = h;
  }
}

// ----------------------------- final linear --------------------------------
// emb[n] = [outh[n] (32), weather[n%8,-1] (8), tenc[n%8,-1] (8)] @ Wl + bl
__global__ __launch_bounds__(64)
void head_kernel(const float* __restrict__ outh,
                 const float* __restrict__ weather,  // [8][12][8]
                 const float* __restrict__ tenc,     // [8][12][8]
                 const float* __restrict__ Wl,       // [48]
                 const float* __restrict__ bl,       // [1]
                 float* __restrict__ pred) {         // [NG*NN]
  int n = blockIdx.x * 64 + threadIdx.x;
  if (n >= NG * NN) return;
  float s = bl[0];
#pragma unroll
  for (int j = 0; j < 32; ++j) s += outh[(size_t)n * 32 + j] * Wl[j];
  int bw = n & 7;  // jnp.tile repeats the 8-row block
#pragma unroll
  for (int f = 0; f < 8; ++f) s += weather[(size_t)bw * TT * 8 + TSTEP * 8 + f] * Wl[32 + f];
#pragma unroll
  for (int f = 0; f < 8; ++f) s += tenc[(size_t)bw * TT * 8 + TSTEP * 8 + f] * Wl[40 + f];
  pred[n] = s;
}

// ------------------------------- launcher ----------------------------------
extern "C" void kernel_launch(void* const* d_in, const int* in_sizes, int n_in,
                              void* d_out, int out_size, void* d_ws, size_t ws_size,
                              hipStream_t stream) {
  const float* x    = (const float*)d_in[0];
  const float* lat  = (const float*)d_in[1];
  const float* lon  = (const float*)d_in[2];
  const float* wth  = (const float*)d_in[3];
  const float* tenc = (const float*)d_in[4];
  const float* W1a  = (const float*)d_in[5];
  const float* b1a  = (const float*)d_in[6];
  const float* W1b  = (const float*)d_in[7];
  const float* b1b  = (const float*)d_in[8];
  const float* W2a  = (const float*)d_in[9];
  const float* b2a  = (const float*)d_in[10];
  const float* W2b  = (const float*)d_in[11];
  const float* b2b  = (const float*)d_in[12];
  const float* W3a  = (const float*)d_in[13];
  const float* b3a  = (const float*)d_in[14];
  const float* W3b  = (const float*)d_in[15];
  const float* b3b  = (const float*)d_in[16];
  const float* Wih  = (const float*)d_in[17];
  const float* Whh  = (const float*)d_in[18];
  const float* bih  = (const float*)d_in[19];
  const float* bhh  = (const float*)d_in[20];
  const float* Wl   = (const float*)d_in[21];
  const float* bl   = (const float*)d_in[22];
  float* pred = (float*)d_out;

  // workspace carve-up (floats)
  float* ws     = (float*)d_ws;
  float* feats0 = ws;                       // NG*NN*3   = 12000
  float* h1     = feats0 + NG * NN * 3;     // NG*NN*32  = 128000
  float* h2     = h1 + NG * NN * 32;        // NG*NN*64  = 256000
  float* h3     = h2 + NG * NN * 64;        // NG*NN*8   = 32000
  float* outh   = h3 + NG * NN * 8;         // NG*NN*32  = 128000
  int*   knn    = (int*)(outh + NG * NN * 32);  // NG*NN*20 ints

  const int NT   = NG * NN;          // 4000
  const int BLK  = (NT + 63) / 64;   // 63
  dim3 ecg(32, NG);                  // 32 node tiles x 8 graphs, 1 wave each

  gather_feats_kernel<<<BLK, 64, 0, stream>>>(x, lat, lon, feats0);

  knn_kernel<3><<<BLK, 64, 0, stream>>>(feats0, knn);
  edgeconv_kernel<3, 12, 32><<<ecg, 32, 0, stream>>>(feats0, knn, W1a, b1a, W1b, b1b, h1);

  knn_kernel<32><<<BLK, 64, 0, stream>>>(h1, knn);
  edgeconv_kernel<32, 64, 64><<<ecg, 32, 0, stream>>>(h1, knn, W2a, b2a, W2b, b2b, h2);

  knn_kernel<64><<<BLK, 64, 0, stream>>>(h2, knn);
  edgeconv_kernel<64, 32, 8><<<ecg, 32, 0, stream>>>(h2, knn, W3a, b3a, W3b, b3b, h3);

  lstm_kernel<<<1, 32, 0, stream>>>(h3, Wih, Whh, bih, bhh, outh);

  head_kernel<<<BLK, 64, 0, stream>>>(outh, wth, tenc, Wl, bl, pred);
}